// HeteroGATLayer_11828339933800
// MI455X (gfx1250) — compile-verified
//
#include <hip/hip_runtime.h>
#include <float.h>

// ---------------------------------------------------------------------------
// HeteroGAT layer for gfx1250 (MI455X). f32 WMMA 16x16x4 for both GEMMs
// (register-blocked: one wave = 16 x (16*NT) output, A-fragment reused across
// NT WMMAs per K-step), native f32 global atomics for edge-softmax phases.
// ---------------------------------------------------------------------------

#define D_INF   128   // input feature dim == H*F
#define H_HEADS 4
#define F_OUT   32
#define T_ET    4
#define NEG_SLOPE 0.2f

typedef __attribute__((ext_vector_type(2))) float v2f;
typedef __attribute__((ext_vector_type(8))) float v8f;

// float atomic max via signed/unsigned int trick (valid when init = -FLT_MAX)
__device__ __forceinline__ void atomicMaxF32(float* addr, float val) {
  if (val >= 0.0f) {
    atomicMax((int*)addr, __float_as_int(val));
  } else {
    atomicMin((unsigned int*)addr, (unsigned int)__float_as_int(val));
  }
}

// ---------------------------------------------------------------------------
// C[M x NN] (+)= A[M x K] @ B[K x NN], row-major. One wave per 16x(16*NT)
// slab: A frag (v2f, 8B load) shared across NT WMMAs per K-step.
// V_WMMA_F32_16X16X4_F32 layouts:
//   A 16x4: lanes 0-15 hold K=0,1 ; lanes 16-31 hold K=2,3 ; M = lane&15
//   B 4x16: mirrored (row K striped across lanes, N = lane&15)
//   C/D 16x16: VGPR r -> M = r + 8*(lane>=16), N = lane&15
// ---------------------------------------------------------------------------
template <int NT>
__global__ void wmma_gemm_f32(const float* __restrict__ A, int lda,
                              const float* __restrict__ B, int ldb,
                              float* __restrict__ C, int ldc,
                              int M, int K, int accumulate) {
  const int mtile  = blockIdx.x;
  const int ntile0 = blockIdx.y * NT;
  const int lane   = threadIdx.x & 31;
  const int half   = lane >> 4;     // 0: lanes 0-15, 1: lanes 16-31
  const int l15    = lane & 15;

  const int arow  = mtile * 16 + l15;
  const int arowc = arow < M ? arow : (M - 1);   // clamp, keep EXEC all-ones

  v8f acc[NT];
#pragma unroll
  for (int nt = 0; nt < NT; ++nt) acc[nt] = (v8f){};

  if (accumulate) {
#pragma unroll
    for (int nt = 0; nt < NT; ++nt) {
#pragma unroll
      for (int r = 0; r < 8; ++r) {
        int mr  = mtile * 16 + r + half * 8;
        int mrc = mr < M ? mr : (M - 1);
        acc[nt][r] = C[(size_t)mrc * ldc + (ntile0 + nt) * 16 + l15];
      }
    }
  }

  // hoisted base pointers; K-loop induction is a single offset add
  const float* ap = A + (size_t)arowc * lda + half * 2;        // A[row][k+half*2]
  const float* bp = B + (size_t)(half * 2) * ldb + ntile0 * 16 + l15;

#pragma unroll 2
  for (int k = 0; k < K; k += 4) {
    v2f a = *(const v2f*)(ap + k);          // 8B contiguous (kk even, lda even)
    const float* bk = bp + (size_t)k * ldb;
#pragma unroll
    for (int nt = 0; nt < NT; ++nt) {
      v2f b;
      b.x = bk[nt * 16];
      b.y = bk[nt * 16 + ldb];
      acc[nt] = __builtin_amdgcn_wmma_f32_16x16x4_f32(
          /*neg_a=*/false, a, /*neg_b=*/false, b,
          /*c_mod=*/(short)0, acc[nt], /*reuse_a=*/false, /*reuse_b=*/false);
    }
  }

#pragma unroll
  for (int nt = 0; nt < NT; ++nt) {
#pragma unroll
    for (int r = 0; r < 8; ++r) {
      int mr = mtile * 16 + r + half * 8;
      if (mr < M) C[(size_t)mr * ldc + (ntile0 + nt) * 16 + l15] = acc[nt][r];
    }
  }
}

// el[n,h] = <ft[n,h,:], attn_l[h,:]> ; er likewise. One thread per (n,h).
__global__ void attn_dots(const float* __restrict__ ft,
                          const float* __restrict__ al,
                          const float* __restrict__ ar,
                          float* __restrict__ el, float* __restrict__ er,
                          int Nn) {
  int idx = blockIdx.x * blockDim.x + threadIdx.x;
  if (idx >= Nn * H_HEADS) return;
  int h = idx & (H_HEADS - 1);
  int n = idx >> 2;
  const float4* f = (const float4*)(ft + (size_t)n * D_INF + h * F_OUT);
  const float4* a = (const float4*)(al + h * F_OUT);
  const float4* b = (const float4*)(ar + h * F_OUT);
  float sl = 0.0f, sr = 0.0f;
#pragma unroll
  for (int i = 0; i < F_OUT / 4; ++i) {
    float4 v  = f[i];
    float4 va = a[i];
    float4 vb = b[i];
    sl += v.x * va.x + v.y * va.y + v.z * va.z + v.w * va.w;
    sr += v.x * vb.x + v.y * vb.y + v.z * vb.z + v.w * vb.w;
  }
  el[idx] = sl;
  er[idx] = sr;
}

__global__ void init_neg_max(float* __restrict__ p, int n) {
  int i = blockIdx.x * blockDim.x + threadIdx.x;
  if (i < n) p[i] = -FLT_MAX;
}

// pass 1: e = leaky(el[src]+er[dst]); store; atomic segment max over dst
__global__ void edge_max_k(const float* __restrict__ el,
                           const float* __restrict__ er,
                           const int* __restrict__ src,
                           const int* __restrict__ dst,
                           float* __restrict__ ew, float* __restrict__ m,
                           int Ecnt) {
  int e = blockIdx.x * blockDim.x + threadIdx.x;
  if (e >= Ecnt) return;
  int s = src[e], d = dst[e];
  float4 l = ((const float4*)el)[s];
  float4 r = ((const float4*)er)[d];
  float v[4] = {l.x + r.x, l.y + r.y, l.z + r.z, l.w + r.w};
#pragma unroll
  for (int h = 0; h < 4; ++h) {
    float x = v[h];
    x = x > 0.0f ? x : NEG_SLOPE * x;
    ew[(size_t)e * 4 + h] = x;
    atomicMaxF32(&m[(size_t)d * 4 + h], x);
  }
}

// pass 2: ex = exp(e - m[dst]); store back; atomic segment sum into denom
__global__ void edge_expsum_k(float* __restrict__ ew,
                              const float* __restrict__ m,
                              const int* __restrict__ dst,
                              float* __restrict__ denom, int Ecnt) {
  int e = blockIdx.x * blockDim.x + threadIdx.x;
  if (e >= Ecnt) return;
  int d = dst[e];
  float4 v  = ((float4*)ew)[e];
  float4 mm = ((const float4*)m)[d];
  v.x = __expf(v.x - mm.x);
  v.y = __expf(v.y - mm.y);
  v.z = __expf(v.z - mm.z);
  v.w = __expf(v.w - mm.w);
  ((float4*)ew)[e] = v;
  atomicAdd(&denom[(size_t)d * 4 + 0], v.x);
  atomicAdd(&denom[(size_t)d * 4 + 1], v.y);
  atomicAdd(&denom[(size_t)d * 4 + 2], v.z);
  atomicAdd(&denom[(size_t)d * 4 + 3], v.w);
}

// pass 3: out[dst] += ft[src] * a  (one wave per edge, lane per feature col)
__global__ void edge_scatter_k(const float* __restrict__ ft,
                               const float* __restrict__ ew,
                               const float* __restrict__ denom,
                               const int* __restrict__ src,
                               const int* __restrict__ dst,
                               float* __restrict__ outp, int Ecnt) {
  int wid  = (int)((blockIdx.x * blockDim.x + threadIdx.x) >> 5);
  int lane = threadIdx.x & 31;
  if (wid >= Ecnt) return;
  int s = src[wid], d = dst[wid];
#pragma unroll
  for (int h = 0; h < 4; ++h) {
    float a = ew[(size_t)wid * 4 + h] / denom[(size_t)d * 4 + h];
    int c = h * F_OUT + lane;
    atomicAdd(&outp[(size_t)d * D_INF + c], ft[(size_t)s * D_INF + c] * a);
  }
}

// final: out = (is_b ? acc_b : acc_a) + b_merge
__global__ void final_select_k(const float* __restrict__ acc_a,
                               const float* __restrict__ acc_b,
                               const int* __restrict__ is_b,
                               const float* __restrict__ bias,
                               float* __restrict__ out, int Nn) {
  int idx = blockIdx.x * blockDim.x + threadIdx.x;
  if (idx >= Nn * F_OUT) return;
  int n = idx >> 5;
  int f = idx & 31;
  const float* p = is_b[n] ? acc_b : acc_a;
  out[idx] = p[idx] + bias[f];
}

// ---------------------------------------------------------------------------
extern "C" void kernel_launch(void* const* d_in, const int* in_sizes, int n_in,
                              void* d_out, int out_size, void* d_ws, size_t ws_size,
                              hipStream_t stream) {
  const float* feat    = (const float*)d_in[0];   // [N,128]
  const int*   src     = (const int*)d_in[1];     // [T,E]
  const int*   dst     = (const int*)d_in[2];     // [T,E]
  const int*   is_b    = (const int*)d_in[3];     // [N]
  const float* W       = (const float*)d_in[4];   // [T,128,128]
  const float* attn_l  = (const float*)d_in[5];   // [T,4,32]
  const float* attn_r  = (const float*)d_in[6];   // [T,4,32]
  const float* W_merge = (const float*)d_in[7];   // [256,32]
  const float* b_merge = (const float*)d_in[8];   // [32]
  float*       out     = (float*)d_out;           // [N,32]

  const int N = in_sizes[0] / D_INF;
  const int E = in_sizes[1] / T_ET;

  // workspace layout (floats)
  float* ws      = (float*)d_ws;
  float* ft      = ws;                 ws += (size_t)N * D_INF;   // [N,128]
  float* out_acc = ws;                 ws += (size_t)N * D_INF;   // [N,128]
  float* el      = ws;                 ws += (size_t)N * H_HEADS; // [N,4]
  float* er      = ws;                 ws += (size_t)N * H_HEADS;
  float* mmax    = ws;                 ws += (size_t)N * H_HEADS;
  float* denom   = ws;                 ws += (size_t)N * H_HEADS;
  float* ew      = ws;                 ws += (size_t)E * H_HEADS; // [E,4]
  float* acc_a   = ws;                 ws += (size_t)N * F_OUT;   // [N,32]
  float* acc_b   = ws;                 ws += (size_t)N * F_OUT;

  const int mtiles = (N + 15) / 16;
  const int eblk   = (E + 255) / 256;
  const int nhblk  = (N * H_HEADS + 255) / 256;
  const int wvblk  = (E * 32 + 255) / 256;   // one wave (32 lanes) per edge
  const int outblk = (N * F_OUT + 255) / 256;

  for (int t = 0; t < T_ET; ++t) {
    hipMemsetAsync(out_acc, 0, (size_t)N * D_INF * sizeof(float), stream);
    hipMemsetAsync(denom,  0, (size_t)N * H_HEADS * sizeof(float), stream);
    init_neg_max<<<nhblk, 256, 0, stream>>>(mmax, N * H_HEADS);

    // ft = feat @ W[t]   (M=N, K=128, NN=128 -> 8 n-tiles, NT=4 per wave)
    wmma_gemm_f32<4><<<dim3(mtiles, 2), 32, 0, stream>>>(
        feat, D_INF, W + (size_t)t * D_INF * D_INF, D_INF,
        ft, D_INF, N, D_INF, /*accumulate=*/0);

    attn_dots<<<nhblk, 256, 0, stream>>>(
        ft, attn_l + t * H_HEADS * F_OUT, attn_r + t * H_HEADS * F_OUT,
        el, er, N);

    const int* src_t = src + (size_t)t * E;
    const int* dst_t = dst + (size_t)t * E;

    edge_max_k   <<<eblk, 256, 0, stream>>>(el, er, src_t, dst_t, ew, mmax, E);
    edge_expsum_k<<<eblk, 256, 0, stream>>>(ew, mmax, dst_t, denom, E);
    edge_scatter_k<<<wvblk, 256, 0, stream>>>(ft, ew, denom, src_t, dst_t,
                                              out_acc, E);

    // acc_{a|b} (+)= out_acc @ W_merge[slice]  (NN=32 -> 2 n-tiles, NT=2)
    float* target = (t < 2) ? acc_a : acc_b;
    const float* Bm = W_merge + (size_t)(t & 1) * D_INF * F_OUT;
    wmma_gemm_f32<2><<<dim3(mtiles, 1), 32, 0, stream>>>(
        out_acc, D_INF, Bm, F_OUT, target, F_OUT, N, D_INF,
        /*accumulate=*/(t & 1));
  }

  final_select_k<<<outblk, 256, 0, stream>>>(acc_a, acc_b, is_b, b_merge,
                                             out, N);
}